// SoftRasterizer3D_67851893342382
// MI455X (gfx1250) — compile-verified
//
#include <hip/hip_runtime.h>
#include <math.h>

// ---------------- problem constants (from reference) ----------------
#define IMG_S   128
#define NPIX    (IMG_S * IMG_S)     // 16384
#define NB      2
#define NV      512
#define NF      512
#define EPSF    1e-8f
#define INV_SIGMA 1e4f              // 1/SIGMA, SIGMA = 1e-4
#define LOG2E   1.44269504088896340736f

typedef __attribute__((ext_vector_type(2))) float v2f;
typedef __attribute__((ext_vector_type(8))) float v8f;
typedef __attribute__((ext_vector_type(4))) int   v4i;

// =====================================================================
// Kernel 1: build view-proj matrix, project all 1024 vertices with
// V_WMMA_F32_16X16X4_F32 (clip = verts_h @ vp^T), perspective-divide,
// then gather per-face triangle xy into the workspace for the raster
// kernel.  One block of 128 threads (4 waves).
// =====================================================================
__global__ void __launch_bounds__(128)
sr_setup_kernel(const float* __restrict__ verts,   // (2,512,3)
                const float* __restrict__ eye,     // (3,)
                const int*   __restrict__ faces,   // (512,3)
                float*       __restrict__ tri)     // out: (2,512,6) a.xy b.xy c.xy
{
    __shared__ float sVP[16];          // vp row-major: vp[r*4+c]
    __shared__ float sD[4][64];        // per-wave 16x4 clip scratch: [m*4+c]
    __shared__ float sNDC[NB * NV * 2];

    const int tid = threadIdx.x;

    // ---- thread 0: look_at + perspective + vp = proj @ view ----
    if (tid == 0) {
        float ex = eye[0], ey = eye[1], ez = eye[2];
        // f = normalize(center - eye), center = 0
        float fx = -ex, fy = -ey, fz = -ez;
        float fn = sqrtf(fx*fx + fy*fy + fz*fz) + EPSF;
        fx /= fn; fy /= fn; fz /= fn;
        // u = up / (|up| + eps), up = (0,1,0)
        float ux = 0.0f, uy = 1.0f / (1.0f + EPSF), uz = 0.0f;
        // s = normalize(cross(f, u))
        float sx = fy*uz - fz*uy;
        float sy = fz*ux - fx*uz;
        float sz = fx*uy - fy*ux;
        float sn = sqrtf(sx*sx + sy*sy + sz*sz) + EPSF;
        sx /= sn; sy /= sn; sz /= sn;
        // u2 = cross(s, f)
        float u2x = sy*fz - sz*fy;
        float u2y = sz*fx - sx*fz;
        float u2z = sx*fy - sy*fx;
        // view = [ s ; u2 ; -f | t ],  t = -(R @ eye)
        float view[16];
        view[0]=sx;   view[1]=sy;   view[2]=sz;   view[3]  = -(sx*ex  + sy*ey  + sz*ez);
        view[4]=u2x;  view[5]=u2y;  view[6]=u2z;  view[7]  = -(u2x*ex + u2y*ey + u2z*ez);
        view[8]=-fx;  view[9]=-fy;  view[10]=-fz; view[11] = -(-fx*ex - fy*ey - fz*ez);
        view[12]=0;   view[13]=0;   view[14]=0;   view[15] = 1.0f;
        // perspective: f=1/tan(22.5deg), M22=(F+N)/(N-F), M23=2FN/(N-F), M32=-1
        const float PF  = 2.4142135623730951f;
        const float M22 = -1.0202020202020203f;
        const float M23 = -0.20202020202020202f;
        // vp = proj @ view (proj is sparse)
        for (int j = 0; j < 4; ++j) {
            sVP[0*4 + j] =  PF  * view[0*4 + j];
            sVP[1*4 + j] =  PF  * view[1*4 + j];
            sVP[2*4 + j] =  M22 * view[2*4 + j] + M23 * view[3*4 + j];
            sVP[3*4 + j] = -view[2*4 + j];
        }
    }
    __syncthreads();

    const int wave = tid >> 5;
    const int lane = tid & 31;
    const int half = lane >> 4;      // 0: K=0,1   1: K=2,3
    const int l    = lane & 15;

    // ---- B operand (4x16 f32): B[k][n] = vp[n][k] for n<4, else 0 ----
    v2f bm; bm.x = 0.0f; bm.y = 0.0f;
    if (l < 4) {
        bm.x = sVP[l*4 + (half ? 2 : 0)];   // K = 0 or 2
        bm.y = sVP[l*4 + (half ? 3 : 1)];   // K = 1 or 3
    }

    // ---- 16 verts / wave / iter; 4 waves; 16 iters -> 1024 verts ----
    for (int it = 0; it < 16; ++it) {
        const int vbase = (it * 4 + wave) * 16;     // [0,1024) step 16
        const int m     = vbase + l;                // this lane's vertex row
        const float* v3 = verts + m * 3;

        // A operand (16x4 f32): row m = (x,y,z,1).
        // Unconditional loads + v_cndmask selects -> uniform control flow
        // (avoids per-iteration saveexec/branch around divergent loads).
        float vx = v3[0];
        float vy = v3[1];
        float vz = v3[2];
        v2f am;
        am.x = half ? vz   : vx;        // K=0 (lanes 0-15) / K=2 (16-31)
        am.y = half ? 1.0f : vy;        // K=1 / K=3

        v8f c = {};
        c = __builtin_amdgcn_wmma_f32_16x16x4_f32(
                /*neg_a=*/false, am, /*neg_b=*/false, bm,
                /*c_mod=*/(short)0, c, /*reuse_a=*/false, /*reuse_b=*/false);

        // D: lanes 0-15 hold col n=l rows 0..7; lanes 16-31 rows 8..15.
        // We only need clip columns 0..3.
        if (l < 4) {
            #pragma unroll
            for (int r = 0; r < 8; ++r)
                sD[wave][(r + half*8)*4 + l] = c[r];
        }
        __syncthreads();

        if (half == 0) {        // lanes 0..15: finish vertex vbase+l
            float cxv = sD[wave][l*4 + 0];
            float cyv = sD[wave][l*4 + 1];
            float cwv = sD[wave][l*4 + 3];
            float inv = __builtin_amdgcn_rcpf(cwv + EPSF);
            sNDC[m*2 + 0] = cxv * inv;
            sNDC[m*2 + 1] = cyv * inv;
        }
        __syncthreads();
    }

    // ---- gather per-face triangle xy: (2*512) records of 6 floats ----
    for (int i = tid; i < NB * NF; i += 128) {
        const int b = i >> 9;
        const int f = i & (NF - 1);
        const int i0 = faces[f*3 + 0];
        const int i1 = faces[f*3 + 1];
        const int i2 = faces[f*3 + 2];
        const float* nb = sNDC + b * NV * 2;
        float* o = tri + i * 6;
        o[0] = nb[i0*2 + 0]; o[1] = nb[i0*2 + 1];
        o[2] = nb[i1*2 + 0]; o[3] = nb[i1*2 + 1];
        o[4] = nb[i2*2 + 0]; o[5] = nb[i2*2 + 1];
    }
}

// =====================================================================
// Kernel 2: soft rasterizer.  128 blocks x 256 threads; each block
// stages its batch's 512x6 face floats (12 KB) into LDS (async DMA if
// available), each thread owns one pixel and loops over all faces.
// sqrt is algebraically removed (sigmoid takes dist^2), and
// 1-sigmoid(x) = 1/(1+exp(x)) -> one v_exp + one v_rcp per face.
// =====================================================================
__device__ __forceinline__ void edge_d2w(float px, float py,
                                         float ax, float ay,
                                         float bx, float by,
                                         float& d2, float& w)
{
    float abx = bx - ax, aby = by - ay;
    float apx = px - ax, apy = py - ay;
    float ab2 = abx*abx + aby*aby;
    float dt  = apx*abx + apy*aby;
    float t   = dt * __builtin_amdgcn_rcpf(ab2 + EPSF);
    t = fminf(fmaxf(t, 0.0f), 1.0f);
    float dx = apx - t*abx;
    float dy = apy - t*aby;
    d2 = dx*dx + dy*dy;
    w  = abx*apy - aby*apx;
}

__global__ void __launch_bounds__(256)
sr_raster_kernel(const float* __restrict__ tri,   // (2,512,6)
                 float*       __restrict__ out)   // (2,128,128)
{
    __shared__ float sTri[NF * 6];                // 12 KB

    const int batch = blockIdx.x >> 6;            // 64 tiles per batch
    const int tile  = blockIdx.x & 63;

    // ---- stage face data into LDS ----
#if __has_builtin(__builtin_amdgcn_global_load_async_to_lds_b128)
    {
        // CDNA5 async DMA path: GLOBAL_LOAD_ASYNC_TO_LDS_B128 (ASYNCcnt).
        // Builtin takes generic int4* pointers (per compiler diagnostic).
        v4i* gsrc = (v4i*)(const_cast<float*>(tri) + batch * NF * 6);
        v4i* ldst = (v4i*)sTri;
        #pragma unroll
        for (int i = threadIdx.x; i < (NF * 6) / 4; i += 256)
            __builtin_amdgcn_global_load_async_to_lds_b128(
                gsrc + i, ldst + i, 0, 0);
    #if __has_builtin(__builtin_amdgcn_s_wait_asynccnt)
        __builtin_amdgcn_s_wait_asynccnt(0);
    #else
        asm volatile("s_wait_asynccnt 0x0" ::: "memory");
    #endif
    }
#else
    {
        const float4* src = (const float4*)(tri + batch * NF * 6);
        float4*       dst = (float4*)sTri;
        #pragma unroll
        for (int i = threadIdx.x; i < (NF * 6) / 4; i += 256)
            dst[i] = src[i];
    }
#endif
    __syncthreads();

    const int pix = tile * 256 + threadIdx.x;     // [0, 16384)
    const int pyi = pix >> 7;
    const int pxi = pix & 127;
    const float gx = (pxi + 0.5f) * (2.0f / IMG_S) - 1.0f;
    const float gy = 1.0f - (pyi + 0.5f) * (2.0f / IMG_S);

    float prod = 1.0f;

    #pragma unroll 4
    for (int f = 0; f < NF; ++f) {
        const float* t6 = &sTri[f * 6];
        float ax = t6[0], ay = t6[1];
        float bx = t6[2], by = t6[3];
        float cx = t6[4], cy = t6[5];

        float d20, d21, d22, w0, w1, w2;
        edge_d2w(gx, gy, ax, ay, bx, by, d20, w0);
        edge_d2w(gx, gy, bx, by, cx, cy, d21, w1);
        edge_d2w(gx, gy, cx, cy, ax, ay, d22, w2);

        float d2 = fminf(fminf(d20, d21), d22) + EPSF;   // == dist*dist

        bool pos = (w0 >= 0.0f) & (w1 >= 0.0f) & (w2 >= 0.0f);
        bool neg = (w0 <= 0.0f) & (w1 <= 0.0f) & (w2 <= 0.0f);
        float sgn = (pos | neg) ? 1.0f : -1.0f;

        // 1 - sigmoid(sgn*d2/sigma) = 1/(1+exp(sgn*d2/sigma))
        float xarg = sgn * d2 * INV_SIGMA;
        float e = __builtin_amdgcn_exp2f(xarg * LOG2E);   // inf -> rcp -> 0, ok
        prod *= __builtin_amdgcn_rcpf(1.0f + e);
    }

    out[batch * NPIX + pix] = 1.0f - prod;
}

// =====================================================================
extern "C" void kernel_launch(void* const* d_in, const int* in_sizes, int n_in,
                              void* d_out, int out_size, void* d_ws, size_t ws_size,
                              hipStream_t stream)
{
    const float* verts = (const float*)d_in[0];   // (2,512,3) f32
    const float* eye   = (const float*)d_in[1];   // (3,)      f32
    const int*   faces = (const int*)  d_in[2];   // (512,3)   i32
    float*       out   = (float*)d_out;           // (2,128,128) f32

    float* tri = (float*)d_ws;                    // (2,512,6) = 24 KB scratch

    sr_setup_kernel<<<1, 128, 0, stream>>>(verts, eye, faces, tri);
    sr_raster_kernel<<<NB * (NPIX / 256), 256, 0, stream>>>(tri, out);
}